// LinkSlotAttention_74689481277674
// MI455X (gfx1250) — compile-verified
//
#include <hip/hip_runtime.h>

typedef __attribute__((ext_vector_type(2))) float v2f;
typedef __attribute__((ext_vector_type(8))) float v8f;
typedef __attribute__((ext_vector_type(4))) unsigned int u32x4;
typedef __attribute__((ext_vector_type(8))) int i32x8;
typedef __attribute__((ext_vector_type(4))) int i32x4;

#define Bsz   8
#define Ssz   512
#define Dsz   256
#define M0sz  100000
#define KTOP  8
#define CAP   (M0sz + Ssz * Bsz)      // 104096
#define CHUNK 8192
#define TROWS 32                      // candidate rows staged per block via TDM

// -------------------------------------------------------------------------
// q = x @ W^T + b   (fp32 WMMA 16x16x4; M=4096, N=256, K=256)
// -------------------------------------------------------------------------
__global__ __launch_bounds__(256) void proj_kernel(const float* __restrict__ x,
                                                   const float* __restrict__ W,
                                                   const float* __restrict__ bias,
                                                   float* __restrict__ qout) {
    const int wave = threadIdx.x >> 5;
    const int l    = threadIdx.x & 31;
    const int tile = blockIdx.x * 8 + wave;          // 4096 tiles total
    const int mbase = (tile >> 4) << 4;              // 256 m-tiles
    const int nbase = (tile & 15) << 4;              // 16 n-tiles
    const int half = l >> 4;
    const int lm   = l & 15;

    const float* arow = x + (size_t)(mbase + lm) * Dsz;   // A[m][k], m=l%16
    const float* brow = W + (size_t)(nbase + lm) * Dsz;   // B[k][n]=W[n][k], n=l%16

    v8f c = {};
    for (int k = 0; k < Dsz; k += 4) {
        v2f a = *(const v2f*)(arow + k + 2 * half);       // k = 2*(l/16)+j
        v2f b = *(const v2f*)(brow + k + 2 * half);
        c = __builtin_amdgcn_wmma_f32_16x16x4_f32(false, a, false, b,
                                                  (short)0, c, false, false);
    }
    const int n = nbase + lm;
    const float bv = bias[n];
    for (int r = 0; r < 8; ++r) {
        const int m = mbase + r + 8 * half;
        qout[(size_t)m * Dsz + n] = c[r] + bv;
    }
}

// -------------------------------------------------------------------------
// qmean[s][d] = mean_b q[b][s][d]
// -------------------------------------------------------------------------
__global__ __launch_bounds__(256) void qmean_kernel(const float* __restrict__ q,
                                                    float* __restrict__ qmean) {
    const int s = blockIdx.x;
    const int d = threadIdx.x;
    float acc = 0.f;
    for (int b = 0; b < Bsz; ++b)
        acc += q[((size_t)b * Ssz + s) * Dsz + d];
    qmean[(size_t)s * Dsz + d] = acc * (1.0f / Bsz);
}

// -------------------------------------------------------------------------
// init running top-k lists
// -------------------------------------------------------------------------
__global__ __launch_bounds__(256) void init_topk(float* __restrict__ gv,
                                                 int* __restrict__ gi) {
    const int i = blockIdx.x * 256 + threadIdx.x;
    if (i < Ssz * KTOP) { gv[i] = -3.4e38f; gi[i] = 0; }
}

// -------------------------------------------------------------------------
// TDM-staged scores tile over the contiguous mem0 segment.
// One block: TDM loads TROWS candidate rows (TROWS x 256 fp32 = 32 KB) into
// LDS once; 8 waves compute all 512 qmean rows against them.
// Wave w covers m-tiles {w, w+8, w+16, w+24} x n-tiles {0,1}; one A fragment
// feeds two WMMAs (B fragments come from LDS via ds_load_b64).
// -------------------------------------------------------------------------
__global__ __launch_bounds__(256) void scores_tdm_kernel(const float* __restrict__ qmean,
                                                         const float* __restrict__ mem0,
                                                         float* __restrict__ sc,
                                                         int chunk_base) {
    __shared__ float shB[TROWS * Dsz];               // 32 KB of 320 KB/WGP
    const int jbase = blockIdx.x * TROWS;            // column offset within chunk
    const int cand0 = chunk_base + jbase;            // first mem0 row of tile
    const int wave  = threadIdx.x >> 5;
    const int l     = threadIdx.x & 31;

    if (wave == 0) {
        // Tensor DMA descriptor (CDNA5 ISA): 2D tile,
        // tile_dim0 = 256 elems (4B), tile_dim1 = TROWS rows, stride 256.
        const unsigned lds_addr = (unsigned)(size_t)(&shB[0]);   // addr[31:0] = LDS offset
        const unsigned long long ga =
            (unsigned long long)(size_t)(mem0 + (size_t)cand0 * Dsz);
        u32x4 g0;
        g0[0] = 1u;                                          // count=1 (valid)
        g0[1] = lds_addr;                                    // lds_addr [63:32]
        g0[2] = (unsigned)(ga & 0xFFFFFFFFull);              // global_addr lo
        g0[3] = (unsigned)((ga >> 32) & 0x01FFFFFFull)       // global_addr [56:32]
              | 0x80000000u;                                 // type=2 ("image")
        i32x8 g1;
        g1[0] = 0x00020000;                                  // data_size=2 (4B)
        g1[1] = (int)(Dsz << 16);                            // tensor_dim0 = 256
        g1[2] = (int)(TROWS << 16);                          // tensor_dim1 = TROWS
        g1[3] = (int)(Dsz << 16);                            // tile_dim0 = 256
        g1[4] = TROWS;                                       // tile_dim1 = TROWS
        g1[5] = Dsz;                                         // tensor_dim0_stride
        g1[6] = 0;
        g1[7] = 0;
        i32x4 z4 = {0, 0, 0, 0};                             // 2D: groups 2/3 unused
        i32x8 z8 = {0, 0, 0, 0, 0, 0, 0, 0};
        __builtin_amdgcn_tensor_load_to_lds(g0, g1, z4, z4, z8, 0);
        __builtin_amdgcn_s_wait_tensorcnt(0);
    }
    __syncthreads();                                         // tile visible to all waves

    const int half = l >> 4;
    const int lm   = l & 15;
    const float* b0row = &shB[(size_t)(lm)      * Dsz];      // n-tile 0, n=l%16
    const float* b1row = &shB[(size_t)(16 + lm) * Dsz];      // n-tile 1

    for (int mi = 0; mi < 4; ++mi) {
        const int mbase = (wave + mi * 8) << 4;
        const float* arow = qmean + (size_t)(mbase + lm) * Dsz;
        v8f c0 = {}, c1 = {};
        for (int k = 0; k < Dsz; k += 4) {
            v2f a  = *(const v2f*)(arow  + k + 2 * half);
            v2f b0 = *(const v2f*)(b0row + k + 2 * half);    // ds_load_b64
            v2f b1 = *(const v2f*)(b1row + k + 2 * half);
            c0 = __builtin_amdgcn_wmma_f32_16x16x4_f32(false, a, false, b0,
                                                       (short)0, c0, false, false);
            c1 = __builtin_amdgcn_wmma_f32_16x16x4_f32(false, a, false, b1,
                                                       (short)0, c1, false, false);
        }
        for (int r = 0; r < 8; ++r) {
            const int m = mbase + r + 8 * half;
            sc[(size_t)m * CHUNK + jbase + lm]      = c0[r];
            sc[(size_t)m * CHUNK + jbase + 16 + lm] = c1[r];
        }
    }
}

// -------------------------------------------------------------------------
// Direct-load scores tile (used for the appended segment scattered in q):
// sc[s][j] = dot(qmean[s], mem_row(chunk_base + j)),
// mem_row(r) = mem0[r] if r < M0 else q[(r-M0)%8][(r-M0)/8]
// -------------------------------------------------------------------------
__global__ __launch_bounds__(256) void scores_kernel(const float* __restrict__ qmean,
                                                     const float* __restrict__ mem0,
                                                     const float* __restrict__ q,
                                                     float* __restrict__ sc,
                                                     int chunk_base, int chunk_n) {
    const int wave = threadIdx.x >> 5;
    const int l    = threadIdx.x & 31;
    const int ntiles = chunk_n >> 4;                 // chunk_n is multiple of 16
    const int tile = blockIdx.x * 8 + wave;
    if (tile >= (Ssz / 16) * ntiles) return;         // wave-uniform guard
    const int mtile = tile / ntiles;
    const int ntile = tile - mtile * ntiles;
    const int half = l >> 4;
    const int lm   = l & 15;
    const int mbase = mtile << 4;
    const int jbase = ntile << 4;

    const float* arow = qmean + (size_t)(mbase + lm) * Dsz;
    const int cand = chunk_base + jbase + lm;
    const int r = cand - M0sz;
    const float* brow = (cand < M0sz)
        ? (mem0 + (size_t)cand * Dsz)
        : (q + ((size_t)(r & 7) * Ssz + (size_t)(r >> 3)) * Dsz);

    v8f c = {};
    for (int k = 0; k < Dsz; k += 4) {
        v2f a = *(const v2f*)(arow + k + 2 * half);
        v2f b = *(const v2f*)(brow + k + 2 * half);
        c = __builtin_amdgcn_wmma_f32_16x16x4_f32(false, a, false, b,
                                                  (short)0, c, false, false);
    }
    for (int rr = 0; rr < 8; ++rr) {
        const int m = mbase + rr + 8 * half;
        sc[(size_t)m * CHUNK + jbase + lm] = c[rr];
    }
}

// -------------------------------------------------------------------------
// per-row top-8 over one chunk, merged into running global top-8
// -------------------------------------------------------------------------
__global__ __launch_bounds__(256) void topk_kernel(const float* __restrict__ sc,
                                                   float* __restrict__ gv,
                                                   int* __restrict__ gi,
                                                   int chunk_base, int chunk_n) {
    const int s = blockIdx.x;
    const int t = threadIdx.x;
    __shared__ float lv[256][KTOP];
    __shared__ int   li[256][KTOP];

    const int count = M0sz + s * Bsz;                // validity at step s
    int limit = count - chunk_base;
    if (limit > chunk_n) limit = chunk_n;
    if (limit < 0) limit = 0;

    float v[KTOP]; int id[KTOP];
    for (int k = 0; k < KTOP; ++k) { v[k] = -3.4e38f; id[k] = 0; }

    const float* row = sc + (size_t)s * CHUNK;
    for (int j = t; j < limit; j += 256) {
        const float x = row[j];
        if (x > v[KTOP - 1]) {
            int p = KTOP - 1;
            while (p > 0 && x > v[p - 1]) { v[p] = v[p - 1]; id[p] = id[p - 1]; --p; }
            v[p] = x; id[p] = chunk_base + j;
        }
    }
    for (int k = 0; k < KTOP; ++k) { lv[t][k] = v[k]; li[t][k] = id[k]; }
    __syncthreads();

    // tournament merge of 256 sorted lists
    for (int stride = 128; stride >= 1; stride >>= 1) {
        if (t < stride) {
            float a[KTOP], m[KTOP]; int ai[KTOP], mi[KTOP];
            for (int k = 0; k < KTOP; ++k) { a[k] = lv[t][k]; ai[k] = li[t][k]; }
            int pa = 0, pb = 0;
            for (int k = 0; k < KTOP; ++k) {
                const float bvv = lv[t + stride][pb];
                if (pa < KTOP && a[pa] >= bvv) { m[k] = a[pa]; mi[k] = ai[pa]; ++pa; }
                else                           { m[k] = bvv;   mi[k] = li[t + stride][pb]; ++pb; }
            }
            for (int k = 0; k < KTOP; ++k) { lv[t][k] = m[k]; li[t][k] = mi[k]; }
        }
        __syncthreads();
    }

    if (t == 0) {   // merge with running global list
        float a[KTOP], m[KTOP]; int ai[KTOP], mi[KTOP];
        for (int k = 0; k < KTOP; ++k) { a[k] = gv[s * KTOP + k]; ai[k] = gi[s * KTOP + k]; }
        int pa = 0, pb = 0;
        for (int k = 0; k < KTOP; ++k) {
            if (pa < KTOP && a[pa] >= lv[0][pb]) { m[k] = a[pa]; mi[k] = ai[pa]; ++pa; }
            else                                 { m[k] = lv[0][pb]; mi[k] = li[0][pb]; ++pb; }
        }
        for (int k = 0; k < KTOP; ++k) { gv[s * KTOP + k] = m[k]; gi[s * KTOP + k] = mi[k]; }
    }
}

// -------------------------------------------------------------------------
// retrieved[s][k][:] = mem_row(idx[s][k])
// -------------------------------------------------------------------------
__global__ __launch_bounds__(256) void gather_kernel(const int* __restrict__ gi,
                                                     const float* __restrict__ mem0,
                                                     const float* __restrict__ q,
                                                     float* __restrict__ retr) {
    const int rr = blockIdx.x;                       // s*KTOP + k
    const int idx = gi[rr];
    const int r = idx - M0sz;
    const float* rowp = (idx < M0sz)
        ? (mem0 + (size_t)idx * Dsz)
        : (q + ((size_t)(r & 7) * Ssz + (size_t)(r >> 3)) * Dsz);
    retr[(size_t)rr * Dsz + threadIdx.x] = rowp[threadIdx.x];
}

// -------------------------------------------------------------------------
// attention per timestep s: att[b][k] via WMMA (8x8 valid in 16x16 tile),
// softmax over k, out[b][:] = sum_k w*retrieved[k][:]
// -------------------------------------------------------------------------
__global__ __launch_bounds__(32) void attn_kernel(const float* __restrict__ q,
                                                  const float* __restrict__ retr,
                                                  float* __restrict__ out) {
    const int s = blockIdx.x;
    const int l = threadIdx.x;
    const int half = l >> 4;
    const int lm   = l & 15;
    __shared__ float att[16][16];
    __shared__ float wgt[Bsz][KTOP];

    const float amask = (lm < Bsz)  ? 1.0f : 0.0f;   // pad rows b>=8 with zeros
    const float bmask = (lm < KTOP) ? 1.0f : 0.0f;   // pad cols k>=8 with zeros
    const float* arow = q    + ((size_t)(lm & 7) * Ssz + s) * Dsz;
    const float* brow = retr + ((size_t)s * KTOP + (lm & 7)) * Dsz;

    v8f c = {};
    for (int k = 0; k < Dsz; k += 4) {
        v2f av = *(const v2f*)(arow + k + 2 * half);
        v2f bv = *(const v2f*)(brow + k + 2 * half);
        v2f a; a.x = av.x * amask; a.y = av.y * amask;
        v2f b; b.x = bv.x * bmask; b.y = bv.y * bmask;
        c = __builtin_amdgcn_wmma_f32_16x16x4_f32(false, a, false, b,
                                                  (short)0, c, false, false);
    }
    for (int r = 0; r < 8; ++r)
        att[r + 8 * half][lm] = c[r] * 0.0625f;      // 1/sqrt(256)
    __syncthreads();

    if (l < Bsz) {
        float mx = -3.4e38f;
        for (int k = 0; k < KTOP; ++k) mx = fmaxf(mx, att[l][k]);
        float e[KTOP]; float sum = 0.f;
        for (int k = 0; k < KTOP; ++k) { e[k] = __expf(att[l][k] - mx); sum += e[k]; }
        const float inv = 1.0f / sum;
        for (int k = 0; k < KTOP; ++k) wgt[l][k] = e[k] * inv;
    }
    __syncthreads();

    for (int b = 0; b < Bsz; ++b) {
        for (int d = l; d < Dsz; d += 32) {
            float acc = 0.f;
            for (int k = 0; k < KTOP; ++k)
                acc += wgt[b][k] * retr[((size_t)s * KTOP + k) * Dsz + d];
            out[((size_t)b * Ssz + s) * Dsz + d] = acc;
        }
    }
}

// -------------------------------------------------------------------------
extern "C" void kernel_launch(void* const* d_in, const int* in_sizes, int n_in,
                              void* d_out, int out_size, void* d_ws, size_t ws_size,
                              hipStream_t stream) {
    const float* x    = (const float*)d_in[0];
    const float* W    = (const float*)d_in[1];
    const float* bias = (const float*)d_in[2];
    const float* mem0 = (const float*)d_in[3];
    float* out = (float*)d_out;

    // workspace layout (floats): q | qmean | sc | gv | gi | retrieved  (~26 MB)
    float* q     = (float*)d_ws;                       // 8*512*256     = 1048576
    float* qmean = q + (size_t)Bsz * Ssz * Dsz;        // 512*256       = 131072
    float* sc    = qmean + (size_t)Ssz * Dsz;          // 512*8192      = 4194304
    float* gv    = sc + (size_t)Ssz * CHUNK;           // 512*8         = 4096
    int*   gi    = (int*)(gv + Ssz * KTOP);            // 512*8         = 4096
    float* retr  = (float*)(gi + Ssz * KTOP);          // 512*8*256     = 1048576

    proj_kernel <<<512, 256, 0, stream>>>(x, W, bias, q);
    qmean_kernel<<<Ssz, 256, 0, stream>>>(q, qmean);
    init_topk   <<<(Ssz * KTOP + 255) / 256, 256, 0, stream>>>(gv, gi);

    // mem0 segment (contiguous rows): TDM-staged WMMA chunks
    int base = 0;
    for (; base + CHUNK <= M0sz; base += CHUNK) {      // 12 chunks of 8192
        scores_tdm_kernel<<<CHUNK / TROWS, 256, 0, stream>>>(qmean, mem0, sc, base);
        topk_kernel<<<Ssz, 256, 0, stream>>>(sc, gv, gi, base, CHUNK);
    }
    {   // mem0 tail: 1696 rows = 53 TDM tiles of 32
        const int tail = M0sz - base;
        scores_tdm_kernel<<<tail / TROWS, 256, 0, stream>>>(qmean, mem0, sc, base);
        topk_kernel<<<Ssz, 256, 0, stream>>>(sc, gv, gi, base, tail);
    }
    {   // appended segment (4096 rows scattered inside q): direct-load WMMA
        const int n = CAP - M0sz;                      // 4096
        const int tiles = (Ssz / 16) * (n >> 4);
        scores_kernel<<<(tiles + 7) / 8, 256, 0, stream>>>(qmean, mem0, q, sc, M0sz, n);
        topk_kernel<<<Ssz, 256, 0, stream>>>(sc, gv, gi, M0sz, n);
    }

    gather_kernel<<<Ssz * KTOP, Dsz, 0, stream>>>(gi, mem0, q, retr);
    attn_kernel  <<<Ssz, 32, 0, stream>>>(q, retr, out);
}